// AttentionLayer_29119878267001
// MI455X (gfx1250) — compile-verified
//
#include <hip/hip_runtime.h>
#include <hip/hip_bf16.h>
#include <math.h>

// Problem constants (reference: B=8, L=1024, D=512, C=4)
static constexpr int Bn = 8;
static constexpr int Ln = 1024;
static constexpr int Dn = 512;
static constexpr int Cn = 4;
static constexpr int XS_STRIDE = 516;  // 16x512 f32 strip, padded: bank = 4*row -> conflict-free

typedef float v2f __attribute__((ext_vector_type(2)));
typedef float v8f __attribute__((ext_vector_type(8)));

// D = A(16x4) * B(4x16) + C, all f32, full precision (matches f32 reference)
__device__ __forceinline__ v8f wmma4(v2f a, v2f b, v8f c) {
  return __builtin_amdgcn_wmma_f32_16x16x4_f32(
      /*neg_a=*/false, a, /*neg_b=*/false, b,
      /*c_mod=*/(short)0, c, /*reuse_a=*/false, /*reuse_b=*/false);
}

__device__ __forceinline__ void load_xstrip(const float* __restrict__ xb,
                                            float (*xs)[XS_STRIDE]) {
  // 16 rows x 512 cols, float4 loads, 256 threads -> 8 float4 each
  for (int t = threadIdx.x; t < 16 * (Dn / 4); t += 256) {
    const int r = t >> 7;          // / 128
    const int q = t & 127;         // % 128
    float4 v = ((const float4*)(xb + (size_t)r * Dn))[q];
    xs[r][4 * q + 0] = v.x;
    xs[r][4 * q + 1] = v.y;
    xs[r][4 * q + 2] = v.z;
    xs[r][4 * q + 3] = v.w;
  }
}

// ---------------------------------------------------------------------------
// K1: Y[b,c,l,e] = sum_d X[b,l,d] * Wl[c,d,e]       (mat_l)
// grid (64 i-tiles, C, B), block 256 (8 waves); each wave does 4 e-tiles.
// ---------------------------------------------------------------------------
__global__ void __launch_bounds__(256)
k1_matl(const float* __restrict__ X, const float* __restrict__ Wl,
        float* __restrict__ Y) {
  const int i0   = blockIdx.x * 16;
  const int c    = blockIdx.y;
  const int b    = blockIdx.z;
  const int lane = threadIdx.x & 31;
  const int wave = threadIdx.x >> 5;
  const int lo   = lane & 15;
  const int lh   = lane >> 4;

  __shared__ float xs[16][XS_STRIDE];
  load_xstrip(X + ((size_t)b * Ln + i0) * Dn, xs);
  __syncthreads();

  const float* Wc = Wl + (size_t)c * Dn * Dn;
  float* Ybc = Y + ((size_t)b * Cn + c) * Ln * Dn;

  for (int ee = 0; ee < 4; ++ee) {
    const int e0 = (ee * 8 + wave) * 16;
    v8f acc = {0.f, 0.f, 0.f, 0.f, 0.f, 0.f, 0.f, 0.f};
#pragma unroll 8
    for (int k0 = 0; k0 < Dn; k0 += 4) {
      const int ka = k0 + 2 * lh;
      v2f a = {xs[lo][ka], xs[lo][ka + 1]};
      // B fragment spans two rows of W (stride D) -> two b32 loads, 64B
      // coalesced per half-wave; W_c stays L2-resident (4 MB total).
      v2f bb = {Wc[(size_t)ka * Dn + e0 + lo],
                Wc[(size_t)(ka + 1) * Dn + e0 + lo]};
      acc = wmma4(a, bb, acc);
    }
#pragma unroll
    for (int r = 0; r < 8; ++r) {
      const int i = i0 + r + 8 * lh;
      Ybc[(size_t)i * Dn + e0 + lo] = acc[r];
    }
  }
}

// ---------------------------------------------------------------------------
// K2: s[b,c,i] = sum_j tanh( (X_b @ Y_bc^T)_ij + b_l[c] )
// grid (64 i-tiles, C, B), block 256; 8 waves x 8 j-tiles each; 128 WMMA/tile.
// Dominant kernel (34.4 GFLOP): B fragment = 2 consecutive floats of the
// lane-private Y row -> single b64 load per WMMA.
// ---------------------------------------------------------------------------
__global__ void __launch_bounds__(256)
k2_rowsum(const float* __restrict__ X, const float* __restrict__ Y,
          const float* __restrict__ bl, float* __restrict__ sOut) {
  const int i0   = blockIdx.x * 16;
  const int c    = blockIdx.y;
  const int b    = blockIdx.z;
  const int lane = threadIdx.x & 31;
  const int wave = threadIdx.x >> 5;
  const int lo   = lane & 15;
  const int lh   = lane >> 4;

  __shared__ float xs[16][XS_STRIDE];
  __shared__ float sl[16];
  load_xstrip(X + ((size_t)b * Ln + i0) * Dn, xs);
  if (threadIdx.x < 16) sl[threadIdx.x] = 0.f;
  __syncthreads();

  const float blc = bl[c];
  const float* Ybc = Y + ((size_t)b * Cn + c) * Ln * Dn;

  float rowsum[8] = {0.f, 0.f, 0.f, 0.f, 0.f, 0.f, 0.f, 0.f};

  for (int jj = 0; jj < 8; ++jj) {
    const int j0 = (jj * 8 + wave) * 16;
    // lane-private Y row; ka is always even -> float2 (b64) aligned loads
    const v2f* yrow2 =
        (const v2f*)(Ybc + (size_t)(j0 + lo) * Dn) + lh;  // element pair base
    v8f acc = {0.f, 0.f, 0.f, 0.f, 0.f, 0.f, 0.f, 0.f};
#pragma unroll 8
    for (int k0 = 0; k0 < Dn; k0 += 4) {
      const int ka = k0 + 2 * lh;
      v2f a = {xs[lo][ka], xs[lo][ka + 1]};
      v2f bb = yrow2[k0 >> 1];  // == {yrow[ka], yrow[ka+1]}
      acc = wmma4(a, bb, acc);
    }
#pragma unroll
    for (int r = 0; r < 8; ++r) {
      float v = tanhf(acc[r] + blc);
      // sum across 16 columns of this tile (masks < 16 never cross halves)
      v += __shfl_xor(v, 1);
      v += __shfl_xor(v, 2);
      v += __shfl_xor(v, 4);
      v += __shfl_xor(v, 8);
      rowsum[r] += v;
    }
  }
  if (lo == 0) {
#pragma unroll
    for (int r = 0; r < 8; ++r) atomicAdd(&sl[r + 8 * lh], rowsum[r]);
  }
  __syncthreads();
  if (threadIdx.x < 16)
    sOut[((size_t)b * Cn + c) * Ln + i0 + threadIdx.x] = sl[threadIdx.x];
}

// ---------------------------------------------------------------------------
// K3: score_bar[b,c,l] = sum_e tanh((X@Wv2)_le + b_v[c,e]) * w_v1[c,e] + pk[b,l]
// ---------------------------------------------------------------------------
__global__ void __launch_bounds__(256)
k3_score(const float* __restrict__ X, const float* __restrict__ Wv2,
         const float* __restrict__ bv, const float* __restrict__ wv1,
         const float* __restrict__ padk, float* __restrict__ scOut) {
  const int i0   = blockIdx.x * 16;
  const int c    = blockIdx.y;
  const int b    = blockIdx.z;
  const int lane = threadIdx.x & 31;
  const int wave = threadIdx.x >> 5;
  const int lo   = lane & 15;
  const int lh   = lane >> 4;

  __shared__ float xs[16][XS_STRIDE];
  __shared__ float scl[16];
  load_xstrip(X + ((size_t)b * Ln + i0) * Dn, xs);
  if (threadIdx.x < 16) scl[threadIdx.x] = 0.f;
  __syncthreads();

  const float* Wc = Wv2 + (size_t)c * Dn * Dn;
  float rowacc[8] = {0.f, 0.f, 0.f, 0.f, 0.f, 0.f, 0.f, 0.f};

  for (int ee = 0; ee < 4; ++ee) {
    const int e0 = (ee * 8 + wave) * 16;
    v8f acc = {0.f, 0.f, 0.f, 0.f, 0.f, 0.f, 0.f, 0.f};
#pragma unroll 8
    for (int k0 = 0; k0 < Dn; k0 += 4) {
      const int ka = k0 + 2 * lh;
      v2f a = {xs[lo][ka], xs[lo][ka + 1]};
      v2f bb = {Wc[(size_t)ka * Dn + e0 + lo],
                Wc[(size_t)(ka + 1) * Dn + e0 + lo]};
      acc = wmma4(a, bb, acc);
    }
    const int e = e0 + lo;
    const float bve = bv[(size_t)c * Dn + e];
    const float wve = wv1[(size_t)c * Dn + e];
#pragma unroll
    for (int r = 0; r < 8; ++r) {
      float v = tanhf(acc[r] + bve) * wve;
      v += __shfl_xor(v, 1);
      v += __shfl_xor(v, 2);
      v += __shfl_xor(v, 4);
      v += __shfl_xor(v, 8);
      rowacc[r] += v;
    }
  }
  if (lo == 0) {
#pragma unroll
    for (int r = 0; r < 8; ++r) atomicAdd(&scl[r + 8 * lh], rowacc[r]);
  }
  __syncthreads();
  if (threadIdx.x < 16) {
    const int l = i0 + threadIdx.x;
    scOut[((size_t)b * Cn + c) * Ln + l] =
        scl[threadIdx.x] + padk[(size_t)b * Ln + l];
  }
}

// ---------------------------------------------------------------------------
// K4: softmaxes + pooled; one block per (c, b)
// ---------------------------------------------------------------------------
__device__ __forceinline__ float block_reduce_max(float v, float* red) {
  const int tid = threadIdx.x;
  red[tid] = v;
  __syncthreads();
  for (int s = 128; s > 0; s >>= 1) {
    if (tid < s) red[tid] = fmaxf(red[tid], red[tid + s]);
    __syncthreads();
  }
  float r = red[0];
  __syncthreads();
  return r;
}

__device__ __forceinline__ float block_reduce_sum(float v, float* red) {
  const int tid = threadIdx.x;
  red[tid] = v;
  __syncthreads();
  for (int s = 128; s > 0; s >>= 1) {
    if (tid < s) red[tid] += red[tid + s];
    __syncthreads();
  }
  float r = red[0];
  __syncthreads();
  return r;
}

__global__ void __launch_bounds__(256)
k4_softmax_pool(const float* __restrict__ X, const float* __restrict__ padk,
                const float* __restrict__ sIn, const float* __restrict__ scIn,
                float* __restrict__ aOut, float* __restrict__ abarOut,
                float* __restrict__ pooled) {
  const int c = blockIdx.x;
  const int b = blockIdx.y;
  const int tid = threadIdx.x;
  const size_t row = ((size_t)b * Cn + c) * Ln;

  __shared__ float buf[Ln];
  __shared__ float abar[Ln];
  __shared__ float red[256];

  // a = softmax(pk + s) over L
  float m = -3.402823466e38f;
  for (int l = tid; l < Ln; l += 256) {
    float v = padk[(size_t)b * Ln + l] + sIn[row + l];
    buf[l] = v;
    m = fmaxf(m, v);
  }
  m = block_reduce_max(m, red);
  float sum = 0.f;
  for (int l = tid; l < Ln; l += 256) {
    float e = expf(buf[l] - m);
    buf[l] = e;
    sum += e;
  }
  sum = block_reduce_sum(sum, red);
  float inv = 1.0f / sum;
  for (int l = tid; l < Ln; l += 256) aOut[row + l] = buf[l] * inv;

  // a_bar = softmax(score_bar) over L
  m = -3.402823466e38f;
  for (int l = tid; l < Ln; l += 256) {
    float v = scIn[row + l];
    abar[l] = v;
    m = fmaxf(m, v);
  }
  m = block_reduce_max(m, red);
  sum = 0.f;
  for (int l = tid; l < Ln; l += 256) {
    float e = expf(abar[l] - m);
    abar[l] = e;
    sum += e;
  }
  sum = block_reduce_sum(sum, red);
  inv = 1.0f / sum;
  for (int l = tid; l < Ln; l += 256) {
    float an = abar[l] * inv;
    abar[l] = an;
    abarOut[row + l] = an;
  }
  __syncthreads();

  // pooled[b,c,d] = sum_l a_bar[l] * X[b,l,d]
  for (int d = tid; d < Dn; d += 256) {
    float acc = 0.f;
    for (int l = 0; l < Ln; ++l)
      acc += abar[l] * X[((size_t)b * Ln + l) * Dn + d];
    pooled[((size_t)b * Cn + c) * Dn + d] = acc;
  }
}

// ---------------------------------------------------------------------------
// K5: C_features[b,l,d,c] = a[b,c,l]*x + pooled[b,c,d]*pad_k2[b,l];  [...,4]=x
// ---------------------------------------------------------------------------
__global__ void __launch_bounds__(256)
k5_assemble(const float* __restrict__ X, const float* __restrict__ padk,
            const float* __restrict__ aIn, const float* __restrict__ pooled,
            float* __restrict__ outF) {
  const size_t idx = (size_t)blockIdx.x * 256 + threadIdx.x;  // over B*L*D
  const size_t total = (size_t)Bn * Ln * Dn;
  if (idx >= total) return;
  const int d = (int)(idx % Dn);
  const size_t bl = idx / Dn;
  const int l = (int)(bl % Ln);
  const int b = (int)(bl / Ln);

  const float x = X[idx];
  const float pk2 = (padk[(size_t)b * Ln + l] + 99999.0f) / 99999.0f;
  float* o = outF + idx * 5;
#pragma unroll
  for (int c = 0; c < Cn; ++c) {
    o[c] = aIn[((size_t)b * Cn + c) * Ln + l] * x +
           pooled[((size_t)b * Cn + c) * Dn + d] * pk2;
  }
  o[4] = x;
}

// ---------------------------------------------------------------------------
extern "C" void kernel_launch(void* const* d_in, const int* in_sizes, int n_in,
                              void* d_out, int out_size, void* d_ws, size_t ws_size,
                              hipStream_t stream) {
  const float* X    = (const float*)d_in[0];  // (B,L,D)
  const float* padk = (const float*)d_in[1];  // (B,L,1)
  const float* wl   = (const float*)d_in[2];  // (C,D,D)
  const float* bl   = (const float*)d_in[3];  // (C,)
  const float* wv1  = (const float*)d_in[4];  // (C,D)
  const float* wv2  = (const float*)d_in[5];  // (C,D,D)
  const float* bv   = (const float*)d_in[6];  // (C,D)

  float* out = (float*)d_out;

  // workspace layout
  const size_t y_elems = (size_t)Bn * Cn * Ln * Dn;       // 16M floats = 64MB
  float* Y      = (float*)d_ws;
  float* sBuf   = Y + y_elems;                            // B*C*L
  float* scBuf  = sBuf + (size_t)Bn * Cn * Ln;            // B*C*L
  float* pooled = scBuf + (size_t)Bn * Cn * Ln;           // B*C*D

  // d_out: [C_features (B*L*D*5)] [a (B*C*L)] [a_bar (B*C*L)]
  float* aOut    = out + (size_t)Bn * Ln * Dn * 5;
  float* abarOut = aOut + (size_t)Bn * Cn * Ln;

  dim3 g(Ln / 16, Cn, Bn);
  k1_matl<<<g, 256, 0, stream>>>(X, wl, Y);
  k2_rowsum<<<g, 256, 0, stream>>>(X, Y, bl, sBuf);
  k3_score<<<g, 256, 0, stream>>>(X, wv2, bv, wv1, padk, scBuf);
  k4_softmax_pool<<<dim3(Cn, Bn), 256, 0, stream>>>(X, padk, sBuf, scBuf,
                                                    aOut, abarOut, pooled);
  const size_t total = (size_t)Bn * Ln * Dn;
  k5_assemble<<<(unsigned)((total + 255) / 256), 256, 0, stream>>>(
      X, padk, aOut, pooled, out);
}